// kernel_generated_1_21887153341281
// MI455X (gfx1250) — compile-verified
//
#include <hip/hip_runtime.h>

typedef __attribute__((ext_vector_type(2))) float v2f;
typedef __attribute__((ext_vector_type(8))) float v8f;

#define NCH  24     // input channels (i)
#define MDIM 56     // m
#define NDIM 56     // n
#define KH   9
#define JD   14
#define RR   (KH*JD)   // 126
#define RPAD 128
#define NPAD 64
#define OC   96

#define WSTR 132   // Wlds row stride (96 x 128 data, padded vs 64 banks)
#define USTR 80    // Ulds row stride (128 x 64 data)
#define TSTR 29    // Tlds row stride (56 x 28 data)
#define XSTR 57    // Xlds row stride (24 x 56 data)

// One block = (m, ntile). Computes out[:, m, ntile*16 : ntile*16+16).
__global__ __launch_bounds__(256) void fused_shift_unfold_wmma(
    const float* __restrict__ x,   // (24,56,56)
    const float* __restrict__ w0,  // (96,9,14)
    const float* __restrict__ w1,  // (24,2,14)
    float* __restrict__ out)       // (96,56,56)
{
    __shared__ float Wlds[OC * WSTR];     // A matrix: w0 as [i][r], r padded to 128
    __shared__ float Ulds[RPAD * USTR];   // B matrix: u as [r][n], padded 128x64
    __shared__ float Tlds[MDIM * TSTR];   // T slice for this m: [h][c], c = k*14+j
    __shared__ float Xlds[NCH * XSTR];    // x[i, m, h]
    __shared__ float W1lds[NCH * 28];     // w1[i][c]

    const int tid   = threadIdx.x;
    const int m     = blockIdx.x;   // 0..55
    const int ntile = blockIdx.y;   // 0..3

    // ---- stage w0 -> Wlds[i][r] (zero-pad r to 128) ----
    for (int e = tid; e < OC * RPAD; e += 256) {
        int i = e >> 7, r = e & 127;
        Wlds[i * WSTR + r] = (r < RR) ? w0[i * RR + r] : 0.0f;
    }
    // ---- stage x slice x[i, m, h] ----
    for (int e = tid; e < NCH * NDIM; e += 256) {
        int i = e / NDIM, h = e % NDIM;
        Xlds[i * XSTR + h] = x[(i * MDIM + m) * NDIM + h];
    }
    // ---- stage w1 ----
    for (int e = tid; e < NCH * 28; e += 256) {
        W1lds[e] = w1[e];
    }
    __syncthreads();

    // ---- T[h][c] = sum_i x[i,m,h] * w1[i][c]  (phase-A einsum, this m only) ----
    for (int e = tid; e < MDIM * 28; e += 256) {
        int h = e / 28, c = e % 28;
        float s = 0.0f;
        #pragma unroll
        for (int i = 0; i < NCH; ++i)
            s += Xlds[i * XSTR + h] * W1lds[i * 28 + c];
        Tlds[h * TSTR + c] = s;
    }
    __syncthreads();

    // ---- U[r][n]: unfold(9,pad 4) + roll(+1 on merged 112 axis) + sum over k ----
    // u[kh,n,j] = T0[n+kh-4, j] + T1[((n-1) mod 56)+kh-4, j], zero-padded in h.
    for (int e = tid; e < RPAD * NPAD; e += 256) {
        int r = e >> 6, n = e & 63;
        float v = 0.0f;
        if (r < RR && n < NDIM) {
            int kh = r / JD;
            int j  = r - kh * JD;
            int h0 = n + kh - 4;                  // k=0 term
            int h1 = ((n + 55) % 56) + kh - 4;    // k=1 term (rolled by +1)
            if (h0 >= 0 && h0 < NDIM) v += Tlds[h0 * TSTR + j];
            if (h1 >= 0 && h1 < NDIM) v += Tlds[h1 * TSTR + 14 + j];
        }
        Ulds[r * USTR + n] = v;
    }
    __syncthreads();

    // ---- out tile = W0 (96x126) x U (126x16-col-slice) via V_WMMA_F32_16X16X4_F32 ----
    const int wave = tid >> 5;
    const int lane = tid & 31;
    if (wave < 6) {                       // uniform per-wave -> EXEC all ones inside
        const int mtile = wave;           // 6 tiles of 16 rows cover 96
        const int arow  = mtile * 16 + (lane & 15);
        const int koff  = (lane >> 4) << 1;          // lanes 16-31 hold K+2 / K+3
        const int col   = ntile * 16 + (lane & 15);

        v8f acc = {};
        #pragma unroll
        for (int ks = 0; ks < RPAD; ks += 4) {
            // A 16x4: VGPR0 = K{0|2}, VGPR1 = K{1|3}; contiguous -> one b64 load
            v2f a = *(const v2f*)&Wlds[arow * WSTR + ks + koff];
            v2f b;
            b.x = Ulds[(ks + koff    ) * USTR + col];
            b.y = Ulds[(ks + koff + 1) * USTR + col];
            acc = __builtin_amdgcn_wmma_f32_16x16x4_f32(
                /*neg_a=*/false, a, /*neg_b=*/false, b,
                /*c_mod=*/(short)0, acc, /*reuse_a=*/false, /*reuse_b=*/false);
        }

        // D layout: VGPR v -> M = v (lanes 0-15) or v+8 (lanes 16-31); N = lane&15
        if (col < NDIM) {
            const int i0 = mtile * 16 + ((lane >> 4) << 3);
            #pragma unroll
            for (int v = 0; v < 8; ++v)
                out[(i0 + v) * (MDIM * NDIM) + m * NDIM + col] = acc[v];
        }
    }
}

extern "C" void kernel_launch(void* const* d_in, const int* in_sizes, int n_in,
                              void* d_out, int out_size, void* d_ws, size_t ws_size,
                              hipStream_t stream) {
    const float* x  = (const float*)d_in[0];  // (1,24,56,56)
    const float* w0 = (const float*)d_in[1];  // (96,9,14)
    const float* w1 = (const float*)d_in[2];  // (24,2,14)
    float* out = (float*)d_out;               // (1,96,56,56)
    (void)in_sizes; (void)n_in; (void)out_size; (void)d_ws; (void)ws_size;

    dim3 grid(MDIM, 4);   // 56 m-slices x 4 n-tiles
    fused_shift_unfold_wmma<<<grid, 256, 0, stream>>>(x, w0, w1, out);
}